// CUTSEncoder_3204045603720
// MI455X (gfx1250) — compile-verified
//
#include <hip/hip_runtime.h>

typedef __attribute__((ext_vector_type(2))) float v2f;
typedef __attribute__((ext_vector_type(8))) float v8f;

#define BN_EPS_F 1e-5f

// LDS offset of activation element for flattened filter-K index k:
// k = ci*9 + r*3 + s  ->  ci*180 + r*18 + s   (tile is [CC][10 rows][18 cols])
__host__ __device__ constexpr int boff(int k) {
  return (k / 9) * 180 + ((k % 9) / 3) * 18 + ((k % 9) % 3);
}

// ---------------------------------------------------------------- utilities
__global__ void zero_f32(float* __restrict__ p, int n) {
  int i = blockIdx.x * blockDim.x + threadIdx.x;
  if (i < n) p[i] = 0.f;
}

// ------------------------------------------------- conv3x3 + bias + BN stats
// One block: 16 couts x (8 rows x 16 cols) spatial tile for one batch image.
// 8 waves; each wave computes a 16(cout) x 16(x) tile for one output row via
// V_WMMA_F32_16X16X4_F32 over K = CIN*9 (chunked CC<=32 input channels in LDS).
// For CC%4==0 the K dimension is permuted to (r,s)-plane-major / ci-minor order
// so every LDS operand address is a compile-time immediate off a fixed base:
// the steady-state inner loop is 4x ds_load_b32 + 1x v_wmma, no address VALU.
template <int CIN>
__global__ __launch_bounds__(256) void conv3x3_wmma_pass1(
    const float* __restrict__ zin,    // (8, CIN, 256, 256)
    const float* __restrict__ wglob,  // (Cout, CIN, 3, 3)
    const float* __restrict__ bias,   // (Cout)
    float* __restrict__ yout,         // (8, Cout, 256, 256); unused if !storeY
    float* __restrict__ gsum, float* __restrict__ gssq,  // (Cout) stats
    int Cout, int coutTiles, int storeY) {
  constexpr int K   = CIN * 9;
  constexpr int CC  = (CIN < 32) ? CIN : 32;  // input channels per LDS chunk
  constexpr int KC  = CC * 9;
  constexpr int KCP = (KC + 3) & ~3;          // pad K-chunk to multiple of 4
  constexpr int NCH = CIN / CC;
  constexpr bool PLANAR = (CC % 4 == 0);      // fast K-permuted path

  extern __shared__ float smem[];
  float* in_lds = smem;               // CC * 10 * 18 halo tile
  float* wlds   = in_lds + CC * 180;  // KCP * 16, [k][m] transposed filters
  float* blds   = wlds + KCP * 16;    // 16 biases
  float* bsum   = blds + 16;          // 16 block partial sums
  float* bssq   = bsum + 16;          // 16 block partial sum-of-squares

  const int tid   = threadIdx.x;
  const int b     = blockIdx.z / coutTiles;
  const int ct    = blockIdx.z - b * coutTiles;
  const int cout0 = ct * 16;
  const int x0    = blockIdx.x * 16;
  const int y0    = blockIdx.y * 8;

  const int lane = tid & 31;
  const int wv   = tid >> 5;   // wave id = output row within tile (0..7)
  const int nx   = lane & 15;  // N (x offset) for B/D, M (cout) for A
  const int hi   = lane >> 4;  // half-wave selector -> K pair

  const int aoff    = nx + 32 * hi;         // A lane offset into wlds
  const int rowbase = wv * 18 + nx;         // B lane offset into in_lds
  const int bbase   = rowbase + hi * 360;   // planar path: hi -> ci += 2

  if (tid < 16) {
    blds[tid] = bias[cout0 + tid];
    bsum[tid] = 0.f;
    bssq[tid] = 0.f;
  }

  v8f acc = {0.f, 0.f, 0.f, 0.f, 0.f, 0.f, 0.f, 0.f};
  const float* zb = zin + (size_t)b * CIN * 65536;

  for (int ch = 0; ch < NCH; ++ch) {
    const int ci0 = ch * CC;
    __syncthreads();
    // Stage activation halo tile: rows y0-1..y0+8, cols x0-1..x0+16 (zero pad)
    for (int idx = tid; idx < CC * 180; idx += 256) {
      unsigned u   = (unsigned)idx;
      unsigned ci  = u / 180u;
      unsigned rem = u - ci * 180u;
      unsigned row = rem / 18u;
      unsigned col = rem - row * 18u;
      int gy = y0 - 1 + (int)row;
      int gx = x0 - 1 + (int)col;
      float v = 0.f;
      if (gy >= 0 && gy < 256 && gx >= 0 && gx < 256)
        v = zb[(size_t)(ci0 + (int)ci) * 65536 + gy * 256 + gx];
      in_lds[idx] = v;
    }
    // Stage 16 filters for this K chunk, transposed to [j][m]; in the planar
    // path j is the permuted K index: j = (r*3+s)*CC + ci.
    for (int idx = tid; idx < 16 * KCP; idx += 256) {
      unsigned m = (unsigned)idx / (unsigned)KCP;
      unsigned j = (unsigned)idx - m * (unsigned)KCP;
      float v = 0.f;
      if ((int)j < KC) {
        unsigned ksrc;
        if (PLANAR) {
          unsigned plane = j / (unsigned)CC;
          unsigned ci    = j - plane * (unsigned)CC;
          ksrc = ci * 9u + plane;
        } else {
          ksrc = j;
        }
        v = wglob[(size_t)(cout0 + (int)m) * K + ci0 * 9 + (int)ksrc];
      }
      wlds[j * 16 + m] = v;
    }
    __syncthreads();

    if constexpr (PLANAR) {
      // j = plane*CC + ci ; group of 4 j's stays inside one (r,s) plane.
#pragma unroll
      for (int j0 = 0; j0 < KC; j0 += 4) {
        const int plane = j0 / CC;                       // compile-time
        const int cib   = j0 - plane * CC;               // compile-time
        const int r = plane / 3, s = plane - r * 3;      // compile-time
        const int base = cib * 180 + r * 18 + s;         // compile-time

        float a_x = wlds[j0 * 16 + aoff];        // A: weights (M = cout)
        float a_y = wlds[j0 * 16 + 16 + aoff];
        float b_x = in_lds[bbase + base];        // B: activations (N = x)
        float b_y = in_lds[bbase + base + 180];

        v2f Af = {a_x, a_y};
        v2f Bf = {b_x, b_y};
        acc = __builtin_amdgcn_wmma_f32_16x16x4_f32(false, Af, false, Bf,
                                                    (short)0, acc, false, false);
      }
    } else {
#pragma unroll
      for (int k0 = 0; k0 < KCP; k0 += 4) {
        const int o0 = boff(k0),     o1 = boff(k0 + 1);
        const int o2 = boff(k0 + 2), o3 = boff(k0 + 3);
        const bool q0 = (k0     < KC), q1 = (k0 + 1 < KC);
        const bool q2 = (k0 + 2 < KC), q3 = (k0 + 3 < KC);

        int offx = hi ? o2 : o0;
        int offy = hi ? o3 : o1;
        bool vx  = hi ? q2 : q0;
        bool vy  = hi ? q3 : q1;

        float a_x = wlds[k0 * 16 + aoff];
        float a_y = wlds[k0 * 16 + 16 + aoff];
        float b_x = vx ? in_lds[offx + rowbase] : 0.f;
        float b_y = vy ? in_lds[offy + rowbase] : 0.f;

        v2f Af = {a_x, a_y};
        v2f Bf = {b_x, b_y};
        acc = __builtin_amdgcn_wmma_f32_16x16x4_f32(false, Af, false, Bf,
                                                    (short)0, acc, false, false);
      }
    }
  }

  // Epilogue: add bias, accumulate BN stats, store conv output.
  const int yy = y0 + wv;
#pragma unroll
  for (int i = 0; i < 8; ++i) {
    int col = i + 8 * hi;  // cout within tile (D: vgpr i -> M = i + 8*hi)
    float v = acc[i] + blds[col];
    float s1 = v;
    float s2 = v * v;
#pragma unroll
    for (int off = 1; off < 16; off <<= 1) {  // reduce over 16 x-positions
      s1 += __shfl_xor(s1, off, 32);
      s2 += __shfl_xor(s2, off, 32);
    }
    if (nx == 0) {
      atomicAdd(&bsum[col], s1);
      atomicAdd(&bssq[col], s2);
    }
    if (storeY)
      yout[(((size_t)b * Cout + cout0 + col) * 256 + yy) * 256 + x0 + nx] = v;
  }

  __syncthreads();
  if (tid < 16) {
    atomicAdd(&gsum[cout0 + tid], bsum[tid]);
    atomicAdd(&gssq[cout0 + tid], bssq[tid]);
  }
}

// ------------------------------------------- BN normalize + leaky ReLU pass
__global__ void bn_lrelu_inplace(float* __restrict__ y,
                                 const float* __restrict__ gsum,
                                 const float* __restrict__ gssq,
                                 const float* __restrict__ gamma,
                                 const float* __restrict__ beta, int Cout) {
  size_t idx = (size_t)blockIdx.x * blockDim.x + threadIdx.x;
  size_t total4 = (size_t)8 * Cout * 65536 / 4;
  if (idx >= total4) return;
  int c = (int)((idx >> 14) % (unsigned)Cout);  // (idx*4 / 65536) % Cout
  const float invN = 1.0f / (8.0f * 65536.0f);
  float mean = gsum[c] * invN;
  float var  = gssq[c] * invN - mean * mean;
  float rs = rsqrtf(var + BN_EPS_F);
  float g = gamma[c], bt = beta[c];
  float4 v = reinterpret_cast<float4*>(y)[idx];
  float t;
  t = g * (v.x - mean) * rs + bt; v.x = t > 0.f ? t : 0.01f * t;
  t = g * (v.y - mean) * rs + bt; v.y = t > 0.f ? t : 0.01f * t;
  t = g * (v.z - mean) * rs + bt; v.z = t > 0.f ? t : 0.01f * t;
  t = g * (v.w - mean) * rs + bt; v.w = t > 0.f ? t : 0.01f * t;
  reinterpret_cast<float4*>(y)[idx] = v;
}

// ---------------------------------------------------- raw image patch gather
__global__ void patches_extract(const float* __restrict__ image,
                                const int* __restrict__ axy,
                                const int* __restrict__ pxy,
                                const int* __restrict__ nxy,
                                float* __restrict__ out) {
  int idx = blockIdx.x * blockDim.x + threadIdx.x;
  if (idx >= 3 * 8 * 3 * 49) return;
  int s   = idx / (8 * 3 * 49);
  int rem = idx - s * (8 * 3 * 49);
  int b   = rem / 147;
  int r2  = rem - b * 147;
  int c   = r2 / 49;
  int r3  = r2 - c * 49;
  int py  = r3 / 7;
  int px  = r3 - py * 7;
  const int* xy = (s == 0) ? axy : ((s == 1) ? pxy : nxy);
  int y = xy[b * 64 + 62] - 3 + py;  // patch index 31 of 32, element 0
  int x = xy[b * 64 + 63] - 3 + px;  // element 1
  out[idx] = image[((size_t)b * 3 + c) * 65536 + y * 256 + x];
}

// ------------------- recompute z3 (conv3+BN3+lrelu) at the 8 gather points
__global__ __launch_bounds__(128) void encode_points(
    const float* __restrict__ z2, const float* __restrict__ w3,
    const float* __restrict__ b3, const float* __restrict__ sum3,
    const float* __restrict__ ssq3, const float* __restrict__ g3,
    const float* __restrict__ be3, const int* __restrict__ nxy,
    float* __restrict__ outEnc) {
  int b  = blockIdx.x;
  int co = threadIdx.x;
  int yc = nxy[b * 64 + 62];
  int xc = nxy[b * 64 + 63];
  const float* wrow = w3 + (size_t)co * 576;
  const float* zb   = z2 + (size_t)b * 64 * 65536;
  float acc = 0.f;
  for (int ci = 0; ci < 64; ++ci)
    for (int r = 0; r < 3; ++r) {
      int gy = yc + r - 1;
      for (int s = 0; s < 3; ++s) {
        int gx = xc + s - 1;
        float zv = (gy >= 0 && gy < 256 && gx >= 0 && gx < 256)
                       ? zb[ci * 65536 + gy * 256 + gx] : 0.f;
        acc += zv * wrow[ci * 9 + r * 3 + s];
      }
    }
  acc += b3[co];
  const float invN = 1.0f / (8.0f * 65536.0f);
  float mean = sum3[co] * invN;
  float var  = ssq3[co] * invN - mean * mean;
  float t = g3[co] * (acc - mean) * rsqrtf(var + BN_EPS_F) + be3[co];
  outEnc[b * 128 + co] = t > 0.f ? t : 0.01f * t;
}

// ---------------------------------------------------------------- launcher
extern "C" void kernel_launch(void* const* d_in, const int* in_sizes, int n_in,
                              void* d_out, int out_size, void* d_ws,
                              size_t ws_size, hipStream_t stream) {
  (void)in_sizes; (void)n_in; (void)out_size; (void)ws_size;
  const float* image = (const float*)d_in[0];
  const int*   axy   = (const int*)d_in[1];
  const int*   pxy   = (const int*)d_in[2];
  const int*   nxy   = (const int*)d_in[3];
  const float* w1 = (const float*)d_in[4];
  const float* b1 = (const float*)d_in[5];
  const float* g1 = (const float*)d_in[6];
  const float* e1 = (const float*)d_in[7];
  const float* w2 = (const float*)d_in[8];
  const float* b2 = (const float*)d_in[9];
  const float* g2 = (const float*)d_in[10];
  const float* e2 = (const float*)d_in[11];
  const float* w3 = (const float*)d_in[12];
  const float* b3 = (const float*)d_in[13];
  const float* g3 = (const float*)d_in[14];
  const float* e3 = (const float*)d_in[15];
  float* out = (float*)d_out;

  float* ws   = (float*)d_ws;
  float* sum1 = ws + 0;    float* ssq1 = ws + 32;
  float* sum2 = ws + 64;   float* ssq2 = ws + 128;
  float* sum3 = ws + 192;  float* ssq3 = ws + 320;
  float* buf1 = ws + 512;                       // 8*32*256*256 floats (64 MB)
  float* buf2 = buf1 + (size_t)8 * 32 * 65536;  // 8*64*256*256 floats (128 MB)

  zero_f32<<<1, 512, 0, stream>>>(ws, 512);  // stats must start at zero

  dim3 blk(256);
  // layer 1: image(3ch) -> buf1(32ch)
  {
    size_t smem = ((size_t)3 * 180 + 28 * 16 + 48) * sizeof(float);
    conv3x3_wmma_pass1<3><<<dim3(16, 32, 8 * 2), blk, smem, stream>>>(
        image, w1, b1, buf1, sum1, ssq1, 32, 2, 1);
    size_t n4 = (size_t)8 * 32 * 65536 / 4;
    bn_lrelu_inplace<<<(n4 + 255) / 256, 256, 0, stream>>>(buf1, sum1, ssq1,
                                                           g1, e1, 32);
  }
  // layer 2: buf1(32ch) -> buf2(64ch)
  {
    size_t smem = ((size_t)32 * 180 + 288 * 16 + 48) * sizeof(float);
    conv3x3_wmma_pass1<32><<<dim3(16, 32, 8 * 4), blk, smem, stream>>>(
        buf1, w2, b2, buf2, sum2, ssq2, 64, 4, 1);
    size_t n4 = (size_t)8 * 64 * 65536 / 4;
    bn_lrelu_inplace<<<(n4 + 255) / 256, 256, 0, stream>>>(buf2, sum2, ssq2,
                                                           g2, e2, 64);
  }
  // layer 3: stats only (no 256 MB store; z3 recomputed at 8 points)
  {
    size_t smem = ((size_t)32 * 180 + 288 * 16 + 48) * sizeof(float);
    conv3x3_wmma_pass1<64><<<dim3(16, 32, 8 * 8), blk, smem, stream>>>(
        buf2, w3, b3, nullptr, sum3, ssq3, 128, 8, 0);
  }
  patches_extract<<<(3 * 8 * 3 * 49 + 255) / 256, 256, 0, stream>>>(
      image, axy, pxy, nxy, out);
  encode_points<<<8, 128, 0, stream>>>(buf2, w3, b3, sum3, ssq3, g3, e3, nxy,
                                       out + 3 * 1176);
}